// f_ss_diag_class_75771813036586
// MI455X (gfx1250) — compile-verified
//
#include <hip/hip_runtime.h>

typedef __attribute__((ext_vector_type(2))) float v2f;
typedef __attribute__((ext_vector_type(8))) float v8f;

#define WAVES_PER_BLOCK 8
#define BLOCK_THREADS (WAVES_PER_BLOCK * 32)

// out[t, n] = x[t, n] * A_diag[n] + sum_k B[n, k] * u[t, k]
// One wave32 per 16-row tile; D = A(u 16x8) x B(B^T 8x16) + C(x * A_diag)
// via two chained V_WMMA_F32_16X16X4_F32 (K=0..3, K=4..7). Exact f32.
__global__ __launch_bounds__(BLOCK_THREADS) void f_ss_diag_wmma_kernel(
    const float* __restrict__ x, const float* __restrict__ u,
    const float* __restrict__ Adiag, const float* __restrict__ Bmat,
    float* __restrict__ out, long long numTiles)
{
    const int lane = threadIdx.x & 31;
    const int wave = threadIdx.x >> 5;
    const long long tile = (long long)blockIdx.x * WAVES_PER_BLOCK + wave;
    if (tile >= numTiles) return;           // wave-uniform: EXEC stays all-1s
    const long long t0 = tile * 16;

    const int col  = lane & 15;             // N index (and M index for A rows)
    const int half = lane >> 4;             // 0: lanes 0-15, 1: lanes 16-31

    // --- C accumulator: C[M][N] = x[t0+M][N] * A_diag[N] ---------------------
    // D/C layout: VGPR v, lanes 0-15 -> M=v, lanes 16-31 -> M=v+8; N = lane&15.
    const float ad = Adiag[col];
    const float* __restrict__ xrow = x + (t0 + half * 8) * 16 + col;
    v8f c;
#pragma unroll
    for (int v = 0; v < 8; ++v)
        c[v] = __builtin_nontemporal_load(xrow + v * 16) * ad;

    // --- A matrix (16x4 f32): lane holds row M=col, K = half*2 + {0,1} -------
    const float* __restrict__ urow = u + (t0 + col) * 8 + half * 2;  // 8B aligned
    const v2f a0 = *(const v2f*)(urow);       // K = 0..3 slice
    const v2f a1 = *(const v2f*)(urow + 4);   // K = 4..7 slice

    // --- B matrix (4x16 f32) = B^T slice: Bt[k][n] = Bmat[n*8 + k] -----------
    // Lane holds N=col, K = half*2 + {0,1} (per-VGPR row striping, K low half
    // in lanes 0-15, high half in lanes 16-31).
    const float* __restrict__ brow = Bmat + col * 8 + half * 2;      // 8B aligned
    const v2f b0 = *(const v2f*)(brow);
    const v2f b1 = *(const v2f*)(brow + 4);

    // D = A*B + C, chained over the K=8 reduction (two K=4 WMMAs).
    c = __builtin_amdgcn_wmma_f32_16x16x4_f32(false, a0, false, b0,
                                              (short)0, c, false, false);
    c = __builtin_amdgcn_wmma_f32_16x16x4_f32(false, a1, false, b1,
                                              (short)0, c, false, false);

    // --- Store D (same coalesced pattern as C init) ---------------------------
    float* __restrict__ orow = out + (t0 + half * 8) * 16 + col;
#pragma unroll
    for (int v = 0; v < 8; ++v)
        __builtin_nontemporal_store(c[v], orow + v * 16);
}

extern "C" void kernel_launch(void* const* d_in, const int* in_sizes, int n_in,
                              void* d_out, int out_size, void* d_ws, size_t ws_size,
                              hipStream_t stream) {
    const float* x     = (const float*)d_in[0];  // (T, 16)
    const float* u     = (const float*)d_in[1];  // (T, 8)
    const float* Adiag = (const float*)d_in[2];  // (16,)
    const float* Bmat  = (const float*)d_in[3];  // (16, 8)
    float* out = (float*)d_out;                  // (T, 16)

    const long long T        = (long long)in_sizes[0] / 16;  // N_X = 16
    const long long numTiles = (T + 15) / 16;                // 250,000 for T=4M
    const long long blocks   = (numTiles + WAVES_PER_BLOCK - 1) / WAVES_PER_BLOCK;

    f_ss_diag_wmma_kernel<<<dim3((unsigned)blocks), BLOCK_THREADS, 0, stream>>>(
        x, u, Adiag, Bmat, out, numTiles);
}